// FuzzyHedgeLayer_68350109549140
// MI455X (gfx1250) — compile-verified
//
#include <hip/hip_runtime.h>

// out[u,i] = min_f( x[u,f] ** e[i,f] ),  x in (0,1)
//          = exp2( min_f( e[i,f] * log2(x[u,f]) ) )        (exp2 monotone)
// Tropical (min,*) GEMM-like op: no WMMA shape exists for min-accumulate,
// so the optimal CDNA5 path is a VALU v_mul/v_min inner loop fed by LDS,
// with operand staging via async global->LDS DMA
// (GLOBAL_LOAD_ASYNC_TO_LDS_B128 + s_wait_asynccnt).

#define N_SAMPLES 8192
#define FDIM      128      // in_features (reduction dim)
#define ODIM      128      // out_features
#define BM        32       // x rows per block
#define BLOCK     256      // 8 wave32
#define CHUNK     64       // output columns staged per LDS chunk
#define EPAD      132      // padded f32 stride for E tile (bank-conflict break)

__global__ __launch_bounds__(BLOCK)
void fuzzy_minpow_kernel(const float* __restrict__ x,
                         const float* __restrict__ e,
                         float* __restrict__ out)
{
    __shared__ float Ltile[BM * FDIM];       // 16 KB  : log2(x) tile
    __shared__ float Etile[CHUNK * EPAD];    // ~33.8KB: exponent chunk (padded)

    const int tid  = threadIdx.x;
    const int lane = tid & 31;
    const int wid  = tid >> 5;
    const int row0 = blockIdx.x * BM;

    // ---- Phase A: x tile -> log2 -> LDS (coalesced float4 loads) ----
    {
        const float4* xg = (const float4*)(x + row0 * FDIM);
        #pragma unroll
        for (int k = 0; k < 4; ++k) {
            const int i = tid + k * BLOCK;         // float4 index in 32x128 tile
            float4 v = xg[i];
            float4 lg;
            lg.x = __builtin_amdgcn_logf(v.x);     // v_log_f32 (log2)
            lg.y = __builtin_amdgcn_logf(v.y);
            lg.z = __builtin_amdgcn_logf(v.z);
            lg.w = __builtin_amdgcn_logf(v.w);
            *(float4*)(Ltile + i * 4) = lg;
        }
    }
    __syncthreads();

    const int r0 = wid * 4;                        // this wave's 4 rows

    for (int c = 0; c < ODIM; c += CHUNK) {
        // ---- Phase B: async DMA exponent chunk [CHUNK x FDIM] -> LDS ----
        // 64x128 f32 = 2048 float4 transfers; 8 per thread.
        #pragma unroll
        for (int k = 0; k < 8; ++k) {
            const int i  = tid + k * BLOCK;        // float4 index within chunk
            const int er = i >> 5;                 // chunk row (= output column)
            const int ef = (i & 31) * 4;           // f offset
            const void* gptr = (const void*)(e + (c + er) * FDIM + ef);
            __attribute__((address_space(3))) void* lptr =
                (__attribute__((address_space(3))) void*)(Etile + er * EPAD + ef);
            asm volatile("global_load_async_to_lds_b128 %0, %1, off"
                         :: "v"(lptr), "v"(gptr) : "memory");
        }
        asm volatile("s_wait_asynccnt 0" ::: "memory");
        __syncthreads();

        // ---- Phase C: tropical (min,*) accumulation, 4 rows x 2 cols/lane ----
        float a00 = __builtin_inff(), a10 = __builtin_inff();
        float a20 = __builtin_inff(), a30 = __builtin_inff();
        float a01 = __builtin_inff(), a11 = __builtin_inff();
        float a21 = __builtin_inff(), a31 = __builtin_inff();
        const float* Ec0 = Etile + lane * EPAD;          // column c+lane
        const float* Ec1 = Etile + (lane + 32) * EPAD;   // column c+lane+32
        #pragma unroll 8
        for (int f = 0; f < FDIM; f += 4) {
            const float4 e0 = *(const float4*)(Ec0 + f);
            const float4 e1 = *(const float4*)(Ec1 + f);
            const float4 l0 = *(const float4*)(Ltile + (r0 + 0) * FDIM + f); // broadcast
            const float4 l1 = *(const float4*)(Ltile + (r0 + 1) * FDIM + f);
            const float4 l2 = *(const float4*)(Ltile + (r0 + 2) * FDIM + f);
            const float4 l3 = *(const float4*)(Ltile + (r0 + 3) * FDIM + f);

            a00 = fminf(a00, fminf(fminf(e0.x*l0.x, e0.y*l0.y), fminf(e0.z*l0.z, e0.w*l0.w)));
            a10 = fminf(a10, fminf(fminf(e0.x*l1.x, e0.y*l1.y), fminf(e0.z*l1.z, e0.w*l1.w)));
            a20 = fminf(a20, fminf(fminf(e0.x*l2.x, e0.y*l2.y), fminf(e0.z*l2.z, e0.w*l2.w)));
            a30 = fminf(a30, fminf(fminf(e0.x*l3.x, e0.y*l3.y), fminf(e0.z*l3.z, e0.w*l3.w)));
            a01 = fminf(a01, fminf(fminf(e1.x*l0.x, e1.y*l0.y), fminf(e1.z*l0.z, e1.w*l0.w)));
            a11 = fminf(a11, fminf(fminf(e1.x*l1.x, e1.y*l1.y), fminf(e1.z*l1.z, e1.w*l1.w)));
            a21 = fminf(a21, fminf(fminf(e1.x*l2.x, e1.y*l2.y), fminf(e1.z*l2.z, e1.w*l2.w)));
            a31 = fminf(a31, fminf(fminf(e1.x*l3.x, e1.y*l3.y), fminf(e1.z*l3.z, e1.w*l3.w)));
        }

        // ---- Phase D: exp2 + coalesced b32 stores (lane -> col c+lane, c+lane+32) ----
        float* og = out + (row0 + r0) * ODIM + c + lane;
        og[0 * ODIM]      = __builtin_amdgcn_exp2f(a00);   // v_exp_f32
        og[1 * ODIM]      = __builtin_amdgcn_exp2f(a10);
        og[2 * ODIM]      = __builtin_amdgcn_exp2f(a20);
        og[3 * ODIM]      = __builtin_amdgcn_exp2f(a30);
        og[0 * ODIM + 32] = __builtin_amdgcn_exp2f(a01);
        og[1 * ODIM + 32] = __builtin_amdgcn_exp2f(a11);
        og[2 * ODIM + 32] = __builtin_amdgcn_exp2f(a21);
        og[3 * ODIM + 32] = __builtin_amdgcn_exp2f(a31);

        __syncthreads();   // protect Etile before next chunk's DMA overwrite
    }
}

extern "C" void kernel_launch(void* const* d_in, const int* in_sizes, int n_in,
                              void* d_out, int out_size, void* d_ws, size_t ws_size,
                              hipStream_t stream) {
    const float* x   = (const float*)d_in[0];   // [8192, 128]
    const float* exp = (const float*)d_in[1];   // [128, 128]
    float*       out = (float*)d_out;           // [8192, 128]
    (void)in_sizes; (void)n_in; (void)out_size; (void)d_ws; (void)ws_size;

    dim3 grid(N_SAMPLES / BM);   // 256 blocks
    dim3 block(BLOCK);           // 256 threads = 8 wave32
    fuzzy_minpow_kernel<<<grid, block, 0, stream>>>(x, exp, out);
}